// LocalAttention_13383118094944
// MI455X (gfx1250) — compile-verified
//
#include <hip/hip_runtime.h>
#include <hip/hip_bf16.h>

// ---------------------------------------------------------------------------
// LocalAttention on MI455X (gfx1250, wave32).
// Memory-bound op: enc (256 MB) streamed twice -> ~22us floor @ 23.3 TB/s.
// Dense projections use V_WMMA_F32_16X16X4_F32 (exact f32 parity with ref).
// enc streaming loads are marked non-temporal (256 MB > 192 MB L2, cyclic
// access -> zero LRU reuse; keep L2 for the small reused tensors instead).
// ---------------------------------------------------------------------------

typedef float v2f __attribute__((ext_vector_type(2)));
typedef float v4f __attribute__((ext_vector_type(4)));
typedef float v8f __attribute__((ext_vector_type(8)));

#define N_B   64
#define L_DIM 1024
#define H_DIM 1024
#define DEV_POW_INV (1.0f / 25.0f)

// ---------------------------------------------------------------------------
// Kernel A: qa = output @ W_a          (which==0, B[k][g] = W_a[k*H+g])
//           q2t = tanh(output @ W_p^T) (which==1, B[k][g] = W_p[g*H+k])
// One wave per 16x16 tile, K = 1024 in steps of 4 (f32 WMMA).
// ---------------------------------------------------------------------------
__global__ __launch_bounds__(32)
void proj_gemm_kernel(const float* __restrict__ out_vec,
                      const float* __restrict__ W_a,
                      const float* __restrict__ W_p,
                      float* __restrict__ qa,
                      float* __restrict__ q2t)
{
    const int lane  = threadIdx.x & 31;
    const int m16   = lane & 15;     // M (for A) / N (for B/C) within tile
    const int khalf = lane >> 4;     // K pair select: 0 -> K{0,1}, 1 -> K{2,3}
    const int n0 = blockIdx.x * 16;
    const int m0 = blockIdx.y * 16;
    const int which = blockIdx.z;

    const float* arow = out_vec + (size_t)(m0 + m16) * H_DIM;
    v8f c = {};

    if (which == 0) {
        for (int k = 0; k < H_DIM; k += 4) {
            const int ka = k + khalf * 2;
            v2f a; a.x = arow[ka]; a.y = arow[ka + 1];
            v2f b;
            b.x = W_a[(size_t)ka * H_DIM + n0 + m16];
            b.y = W_a[(size_t)(ka + 1) * H_DIM + n0 + m16];
            c = __builtin_amdgcn_wmma_f32_16x16x4_f32(false, a, false, b,
                                                      (short)0, c, false, false);
        }
    } else {
        const float* brow = W_p + (size_t)(n0 + m16) * H_DIM;  // transposed access
        for (int k = 0; k < H_DIM; k += 4) {
            const int ka = k + khalf * 2;
            v2f a; a.x = arow[ka]; a.y = arow[ka + 1];
            v2f b; b.x = brow[ka]; b.y = brow[ka + 1];
            c = __builtin_amdgcn_wmma_f32_16x16x4_f32(false, a, false, b,
                                                      (short)0, c, false, false);
        }
    }

    // C/D layout: VGPR r -> M = m0 + r + 8*khalf, N = n0 + (lane&15)
    const int nCol  = n0 + m16;
    const int mBase = m0 + khalf * 8;
    if (which == 0) {
#pragma unroll
        for (int r = 0; r < 8; ++r)
            qa[(size_t)(mBase + r) * H_DIM + nCol] = c[r];
    } else {
#pragma unroll
        for (int r = 0; r < 8; ++r)
            q2t[(size_t)(mBase + r) * H_DIM + nCol] = tanhf(c[r]);
    }
}

// ---------------------------------------------------------------------------
// Kernel P: p_t[n] = S * sigmoid( dot(q2t[n,:], v_p) ),  S = H (ref's bug)
// ---------------------------------------------------------------------------
__global__ __launch_bounds__(256)
void pos_kernel(const float* __restrict__ q2t,
                const float* __restrict__ v_p,
                float* __restrict__ p_t)
{
    const int n = blockIdx.x;
    __shared__ float red[256];
    float acc = 0.f;
    for (int h = threadIdx.x; h < H_DIM; h += 256)
        acc += q2t[(size_t)n * H_DIM + h] * v_p[h];
    red[threadIdx.x] = acc;
    __syncthreads();
    for (int s = 128; s > 0; s >>= 1) {
        if (threadIdx.x < s) red[threadIdx.x] += red[threadIdx.x + s];
        __syncthreads();
    }
    if (threadIdx.x == 0) {
        const float z = red[0];
        p_t[n] = (float)H_DIM / (1.f + __expf(-z));
    }
}

// ---------------------------------------------------------------------------
// Kernel B: scores[n,l] = dot(enc[n,l,:], qa[n,:]).  Streams enc (pass 1, NT).
// 8 waves per block, one l-row per wave; qa row staged in LDS; 128-bit loads.
// ---------------------------------------------------------------------------
__global__ __launch_bounds__(256)
void score_kernel(const float* __restrict__ enc,
                  const float* __restrict__ qa,
                  float* __restrict__ scores)
{
    const int n = blockIdx.y;
    __shared__ float q[H_DIM];
    for (int i = threadIdx.x; i < H_DIM; i += 256)
        q[i] = qa[(size_t)n * H_DIM + i];
    __syncthreads();

    const int wave = threadIdx.x >> 5;
    const int lane = threadIdx.x & 31;
    const int l = blockIdx.x * 8 + wave;
    const float* row = enc + ((size_t)n * L_DIM + l) * H_DIM;

    float acc = 0.f;
#pragma unroll
    for (int it = 0; it < 8; ++it) {
        const int h = it * 128 + lane * 4;
        const v4f e = __builtin_nontemporal_load((const v4f*)(row + h));
        acc += e.x * q[h] + e.y * q[h + 1] + e.z * q[h + 2] + e.w * q[h + 3];
    }
#pragma unroll
    for (int m = 16; m >= 1; m >>= 1) acc += __shfl_xor(acc, m, 32);
    if (lane == 0) scores[(size_t)n * L_DIM + l] = acc;
}

// ---------------------------------------------------------------------------
// Kernel C: align[n,l] = softmax_l(scores[n,:]) * exp(-(l - p_t)^2 / 25)
// One block per batch row; 4 elements per thread.
// ---------------------------------------------------------------------------
__global__ __launch_bounds__(256)
void softmax_gauss_kernel(const float* __restrict__ scores,
                          const float* __restrict__ p_t,
                          float* __restrict__ align_)
{
    const int n = blockIdx.x;
    __shared__ float red[256];
    const int base = threadIdx.x * 4;
    const float* s = scores + (size_t)n * L_DIM;

    float v[4];
    float mx = -INFINITY;
#pragma unroll
    for (int j = 0; j < 4; ++j) { v[j] = s[base + j]; mx = fmaxf(mx, v[j]); }
    red[threadIdx.x] = mx;
    __syncthreads();
    for (int st = 128; st > 0; st >>= 1) {
        if (threadIdx.x < st)
            red[threadIdx.x] = fmaxf(red[threadIdx.x], red[threadIdx.x + st]);
        __syncthreads();
    }
    mx = red[0];
    __syncthreads();

    float sum = 0.f;
#pragma unroll
    for (int j = 0; j < 4; ++j) { v[j] = __expf(v[j] - mx); sum += v[j]; }
    red[threadIdx.x] = sum;
    __syncthreads();
    for (int st = 128; st > 0; st >>= 1) {
        if (threadIdx.x < st) red[threadIdx.x] += red[threadIdx.x + st];
        __syncthreads();
    }
    const float inv = 1.f / red[0];
    const float p = p_t[n];

    float* a = align_ + (size_t)n * L_DIM;
#pragma unroll
    for (int j = 0; j < 4; ++j) {
        const float d = (float)(base + j) - p;
        a[base + j] = v[j] * inv * __expf(-(d * d) * DEV_POW_INV);
    }
}

// ---------------------------------------------------------------------------
// Kernel D: context split-K partials.  Streams enc (pass 2, NT).
// Block (lq, n) handles 256 l-rows x all 1024 h-cols (128-bit per thread).
// ---------------------------------------------------------------------------
__global__ __launch_bounds__(256)
void context_partial_kernel(const float* __restrict__ enc,
                            const float* __restrict__ align_,
                            float* __restrict__ partial)
{
    const int n = blockIdx.y, lq = blockIdx.x;
    __shared__ float a[256];
    a[threadIdx.x] = align_[(size_t)n * L_DIM + lq * 256 + threadIdx.x];
    __syncthreads();

    const int h = threadIdx.x * 4;
    v4f acc = {0.f, 0.f, 0.f, 0.f};
    const float* base = enc + ((size_t)n * L_DIM + (size_t)lq * 256) * H_DIM + h;
    for (int l = 0; l < 256; ++l) {
        const v4f e = __builtin_nontemporal_load((const v4f*)(base + (size_t)l * H_DIM));
        const float w = a[l];
        acc.x += w * e.x; acc.y += w * e.y; acc.z += w * e.z; acc.w += w * e.w;
    }
    *(v4f*)(partial + ((size_t)(n * 4 + lq)) * H_DIM + h) = acc;
}

// ---------------------------------------------------------------------------
// Kernel D2: cat[n, 0:H] = sum of 4 context partials; cat[n, H:2H] = output.
// ---------------------------------------------------------------------------
__global__ __launch_bounds__(256)
void cat_kernel(const float* __restrict__ partial,
                const float* __restrict__ out_vec,
                float* __restrict__ cat)
{
    const int idx = blockIdx.x * 256 + threadIdx.x;  // [0, N*2H)
    const int n = idx >> 11;
    const int c = idx & 2047;
    float v;
    if (c < H_DIM) {
        const float* p = partial + (size_t)n * 4 * H_DIM + c;
        v = p[0] + p[H_DIM] + p[2 * H_DIM] + p[3 * H_DIM];
    } else {
        v = out_vec[(size_t)n * H_DIM + (c - H_DIM)];
    }
    cat[idx] = v;
}

// ---------------------------------------------------------------------------
// Kernel E: out = tanh(cat @ W_c^T), M=64, N=1024, K=2048, f32 WMMA.
// B[k][h] = W_c[h*2H + k] (transposed access; W_c is 8 MB -> L2 resident).
// ---------------------------------------------------------------------------
__global__ __launch_bounds__(32)
void final_gemm_kernel(const float* __restrict__ cat,
                       const float* __restrict__ W_c,
                       float* __restrict__ out)
{
    const int K = 2 * H_DIM;
    const int lane  = threadIdx.x & 31;
    const int m16   = lane & 15;
    const int khalf = lane >> 4;
    const int n0 = blockIdx.x * 16;
    const int m0 = blockIdx.y * 16;

    const float* arow = cat + (size_t)(m0 + m16) * K;
    const float* brow = W_c + (size_t)(n0 + m16) * K;
    v8f c = {};
    for (int k = 0; k < K; k += 4) {
        const int ka = k + khalf * 2;
        v2f a; a.x = arow[ka]; a.y = arow[ka + 1];
        v2f b; b.x = brow[ka]; b.y = brow[ka + 1];
        c = __builtin_amdgcn_wmma_f32_16x16x4_f32(false, a, false, b,
                                                  (short)0, c, false, false);
    }
    const int nCol  = n0 + m16;
    const int mBase = m0 + khalf * 8;
#pragma unroll
    for (int r = 0; r < 8; ++r)
        out[(size_t)(mBase + r) * H_DIM + nCol] = tanhf(c[r]);
}

// ---------------------------------------------------------------------------
// Launch
// ---------------------------------------------------------------------------
extern "C" void kernel_launch(void* const* d_in, const int* in_sizes, int n_in,
                              void* d_out, int out_size, void* d_ws, size_t ws_size,
                              hipStream_t stream) {
    const float* enc     = (const float*)d_in[0];  // (64,1024,1024)
    const float* out_vec = (const float*)d_in[1];  // (64,1,1024)
    // d_in[2] = time_step (unused by the reference computation)
    const float* W_a     = (const float*)d_in[3];  // (1024,1024)
    const float* W_p     = (const float*)d_in[4];  // (1024,1024)
    const float* v_p     = (const float*)d_in[5];  // (1,1024)
    const float* W_c     = (const float*)d_in[6];  // (1024,2048)
    float* out = (float*)d_out;

    // Workspace carve-up (floats)
    float* ws      = (float*)d_ws;
    float* qa      = ws;               // 64*1024
    float* q2t     = ws + 65536;       // 64*1024
    float* p_t     = ws + 131072;      // 64
    float* scores  = ws + 131136;      // 64*1024
    float* align_  = ws + 196672;      // 64*1024
    float* partial = ws + 262208;      // 64*4*1024
    float* cat     = ws + 524352;      // 64*2048

    proj_gemm_kernel<<<dim3(H_DIM / 16, N_B / 16, 2), 32, 0, stream>>>(
        out_vec, W_a, W_p, qa, q2t);
    pos_kernel<<<N_B, 256, 0, stream>>>(q2t, v_p, p_t);
    score_kernel<<<dim3(L_DIM / 8, N_B), 256, 0, stream>>>(enc, qa, scores);
    softmax_gauss_kernel<<<N_B, 256, 0, stream>>>(scores, p_t, align_);
    context_partial_kernel<<<dim3(4, N_B), 256, 0, stream>>>(enc, align_, partial);
    cat_kernel<<<(N_B * 2 * H_DIM) / 256, 256, 0, stream>>>(partial, out_vec, cat);
    final_gemm_kernel<<<dim3(H_DIM / 16, N_B / 16), 32, 0, stream>>>(cat, W_c, out);
}